// PigletImageEncoder_73967926772599
// MI455X (gfx1250) — compile-verified
//
#include <hip/hip_runtime.h>

#define HEIGHT 384
#define WIDTH  640
#define NPIX   (HEIGHT * WIDTH)   // 245760
#define NPIX4  (NPIX / 4)         // 61440 float4 per plane
#define NB     8
#define NQ     100
#define DIM    256

typedef __attribute__((ext_vector_type(2))) float v2f;
typedef __attribute__((ext_vector_type(4))) float v4f;
typedef __attribute__((ext_vector_type(8))) float v8f;

// ---------------------------------------------------------------------------
// Kernel 1: per-pair channel-summed abs diff, float4-vectorized.
//   diff[p, pix] = | sum_c img[2p+1,c,pix] - sum_c img[2p,c,pix] |, p = 0..3
// 23.6 MB in / 3.9 MB out -> HBM bound (~1.2 us @ 23.3 TB/s).  b128 loads give
// 7 memory instructions per 16 output bytes and deep load pipelining.
// ---------------------------------------------------------------------------
__global__ void diff_kernel(const float* __restrict__ images,
                            float* __restrict__ diff) {
    int idx = blockIdx.x * blockDim.x + threadIdx.x;   // one float4 per thread
    if (idx >= 4 * NPIX4) return;
    int p    = idx / NPIX4;
    int pix4 = idx - p * NPIX4;
    const v4f* a = (const v4f*)(images + (size_t)(2 * p)     * 3 * NPIX) + pix4;
    const v4f* b = (const v4f*)(images + (size_t)(2 * p + 1) * 3 * NPIX) + pix4;
    v4f sa = a[0] + a[NPIX4] + a[2 * NPIX4];
    v4f sb = b[0] + b[NPIX4] + b[2 * NPIX4];
    v4f d  = sb - sa;
    v4f r;
    r.x = fabsf(d.x); r.y = fabsf(d.y); r.z = fabsf(d.z); r.w = fabsf(d.w);
    ((v4f*)diff)[idx] = r;
}

// ---------------------------------------------------------------------------
// Kernel 2: one workgroup per (b, n) box.  Faithful to the reference's axis
// convention: ROW bounds from the WIDTH-scaled coords (bb0, bb2), COLUMN
// bounds from the HEIGHT-scaled coords (bb1, bb3); inclusive bounds; int
// conversion truncates toward zero.  score = boxSum(diff) / (H*W).
// Threads tiled 4 (rows) x 64 (cols): constant strides, no div/mod in the
// inner loop, lanes sweep consecutive columns -> coalesced L2-resident loads.
// ---------------------------------------------------------------------------
__global__ void score_kernel(const float* __restrict__ pred_boxes,
                             const float* __restrict__ diff,
                             float* __restrict__ scores) {
    int bn = blockIdx.x;              // 0..799
    int b  = bn / NQ;
    const float* box = pred_boxes + (size_t)bn * 4;
    float cx = box[0], cy = box[1], w = box[2], h = box[3];
    int bb0 = (int)((cx - 0.5f * w) * (float)WIDTH);    // trunc == astype(int32)
    int bb1 = (int)((cy - 0.5f * h) * (float)HEIGHT);
    int bb2 = (int)((cx + 0.5f * w) * (float)WIDTH);
    int bb3 = (int)((cy + 0.5f * h) * (float)HEIGHT);
    int r_lo = max(bb0, 0), r_hi = min(bb2, HEIGHT - 1);   // rows:    bb0 <= x <= bb2
    int c_lo = max(bb1, 0), c_hi = min(bb3, WIDTH  - 1);   // columns: bb1 <= y <= bb3

    int tx = threadIdx.x & 63;        // column lane (0..63)
    int ty = threadIdx.x >> 6;        // row lane    (0..3)

    float acc = 0.0f;
    if (r_lo <= r_hi && c_lo <= c_hi) {
        const float* dp = diff + (size_t)(b >> 1) * NPIX;  // repeat(2) on batch
        for (int r = r_lo + ty; r <= r_hi; r += 4) {
            const float* row = dp + r * WIDTH;
            for (int c = c_lo + tx; c <= c_hi; c += 64)
                acc += row[c];
        }
    }
    __shared__ float red[256];
    red[threadIdx.x] = acc;
    __syncthreads();
    for (int s = 128; s > 0; s >>= 1) {
        if ((int)threadIdx.x < s) red[threadIdx.x] += red[threadIdx.x + s];
        __syncthreads();
    }
    if (threadIdx.x == 0) scores[bn] = red[0] * (1.0f / (float)NPIX);
}

// ---------------------------------------------------------------------------
// Kernel 3: fused top-2 -> gather -> Linear (WMMA f32) -> scatter to d_out.
// One workgroup, 8 waves (wave32).  X is [16 x 256] (rows m=(b,o)); Y = X @
// fc_w^T + fc_b is one 16-row WMMA band: each wave owns two 16x16 N-tiles,
// 64 V_WMMA_F32_16X16X4_F32 K-steps per tile (exact f32 accumulate).
//
// 16x4 f32 A fragment (ISA 7.12.2): lanes 0-15 hold {K=kk,kk+1} of row
// M=lane; lanes 16-31 hold {K=kk+2,kk+3} of row M=lane-16.  B (4x16) is the
// transposed analog; since B[k,n] = fc_w[n,k], each lane's B fragment is a
// contiguous 8-byte global load of one fc_w row -> fc_w streamed exactly
// once (256 KB) across the 8 waves.
// ---------------------------------------------------------------------------
__global__ void head_kernel(const float* __restrict__ scores,
                            const float* __restrict__ hidden,
                            const float* __restrict__ fc_w,
                            const float* __restrict__ fc_b,
                            float* __restrict__ out) {
    __shared__ int   idx_sm[16];
    __shared__ float X[16][260];      // stride 260: lane L hits bank 4L (conflict-free)

    int tid = threadIdx.x;

    // --- top-2 per batch (threads 0..7); strict '>' == top_k's stable ties ---
    if (tid < NB) {
        const float* s = scores + tid * NQ;
        int i1 = 0; float v1 = s[0];
        for (int i = 1; i < NQ; i++) { float v = s[i]; if (v > v1) { v1 = v; i1 = i; } }
        int i2 = (i1 == 0) ? 1 : 0; float v2 = s[i2];
        for (int i = 0; i < NQ; i++) {
            if (i == i1) continue;
            float v = s[i];
            if (v > v2) { v2 = v; i2 = i; }
        }
        idx_sm[tid * 2 + 0] = i1;
        idx_sm[tid * 2 + 1] = i2;
    }
    __syncthreads();

    // --- gather X[m, :] = hidden[b, idx, :],  m = 2*b + o ---
    for (int t = tid; t < 16 * DIM; t += 256) {
        int m = t >> 8;           // row
        int k = t & 255;          // column
        int b = m >> 1;
        X[m][k] = hidden[(size_t)b * NQ * DIM + (size_t)idx_sm[m] * DIM + k];
    }
    __syncthreads();

    // --- WMMA GEMM: Y[16,256] = X @ fc_w^T ---
    int wave  = tid >> 5;         // 0..7
    int lane  = tid & 31;
    int lhalf = lane >> 4;        // 0: K lo pair, 1: K hi pair
    int l16   = lane & 15;
    int n0    = wave * 32;        // this wave's two N-tiles: n0, n0+16

    v8f c0 = {}; v8f c1 = {};
    for (int kk = 0; kk < DIM; kk += 4) {
        int ka = kk + 2 * lhalf;
        v2f a;  a.x  = X[l16][ka];  a.y = X[l16][ka + 1];
        const float* w0 = fc_w + (size_t)(n0 +      l16) * DIM + ka;
        const float* w1 = fc_w + (size_t)(n0 + 16 + l16) * DIM + ka;
        v2f b0; b0.x = w0[0]; b0.y = w0[1];
        v2f b1; b1.x = w1[0]; b1.y = w1[1];
        c0 = __builtin_amdgcn_wmma_f32_16x16x4_f32(false, a, false, b0, (short)0, c0, false, false);
        c1 = __builtin_amdgcn_wmma_f32_16x16x4_f32(false, a, false, b1, (short)0, c1, false, false);
    }

    // --- bias + scatter: row m=(b,o) -> out[(b&1)*2048 + (b>>1)*512 + o*256 + n]
    // C/D layout: VGPR v holds M = v + 8*lhalf, N = n0(+16) + l16.
    for (int t = 0; t < 2; t++) {
        v8f c    = t ? c1 : c0;
        int ncol = n0 + 16 * t + l16;
        float bias = fc_b[ncol];
        #pragma unroll
        for (int v = 0; v < 8; v++) {
            int m = v + 8 * lhalf;
            int b = m >> 1, o = m & 1;
            out[(b & 1) * 2048 + (b >> 1) * 512 + o * 256 + ncol] = c[v] + bias;
        }
    }
}

// ---------------------------------------------------------------------------
extern "C" void kernel_launch(void* const* d_in, const int* in_sizes, int n_in,
                              void* d_out, int out_size, void* d_ws, size_t ws_size,
                              hipStream_t stream) {
    const float* images     = (const float*)d_in[0];
    // d_in[1] (logits) is dead code in the reference (softmax result unused)
    const float* pred_boxes = (const float*)d_in[2];
    const float* hidden     = (const float*)d_in[3];
    const float* fc_w       = (const float*)d_in[4];
    const float* fc_b       = (const float*)d_in[5];
    float*       out        = (float*)d_out;

    float* diff   = (float*)d_ws;            // 4 * 245760 f32 = 3.93 MB
    float* scores = diff + 4 * NPIX;         // 800 f32

    diff_kernel <<<(4 * NPIX4 + 255) / 256, 256, 0, stream>>>(images, diff);
    score_kernel<<<NB * NQ,               256, 0, stream>>>(pred_boxes, diff, scores);
    head_kernel <<<1,                     256, 0, stream>>>(scores, hidden, fc_w, fc_b, out);
}